// NUFFTLayerMultiChannel2Dmixed_10093173146140
// MI455X (gfx1250) — compile-verified
//
#include <hip/hip_runtime.h>
#include <math.h>

typedef __attribute__((ext_vector_type(2))) float v2f;
typedef __attribute__((ext_vector_type(8))) float v8f;

#define MM    129            // mesh size (odd)
#define MP    144            // padded to multiple of 16 (9 tiles)
#define NPP   512            // points per batch
#define NB    4              // batches
#define NPTS  (NB * NPP)     // 2048

#define PI_F   3.14159265358979323846f
#define TAU_F  (12.0f / (129.0f * 129.0f))   // tau = 12*(L/(2*pi*M))^2 with L=2*pi
#define L_F    (2.0f * PI_F)

// ---------------- WMMA helpers (f32 16x16x4) ----------------

__device__ __forceinline__ v8f wmma4(v2f a, v2f b, v8f c) {
  // (neg_a, A, neg_b, B, c_mod, C, reuse_a, reuse_b); f32 has no A/B neg -> false
  return __builtin_amdgcn_wmma_f32_16x16x4_f32(false, a, false, b, (short)0, c, false, false);
}

// C/D layout: VGPR v -> row = v + 8*(lane>>4), col = lane&15
__device__ __forceinline__ void store_tile(float* C, int ldc, int row0, int col0, v8f acc) {
  int lane = threadIdx.x & 31;
  int r0 = row0 + ((lane >> 4) << 3);
  int c  = col0 + (lane & 15);
#pragma unroll
  for (int v = 0; v < 8; ++v) C[(r0 + v) * ldc + c] = acc[v];
}

// Generic one-wave 16x16 tile GEMM: acc += op(A)[row0:+16, :K] * op(B)[:K, col0:+16]
// A frag: row = lane&15, K = k0 + 2*(lane>>4) + {0,1}
// B frag: col = lane&15, K = k0 + 2*(lane>>4) + {0,1}
template <typename FA, typename FB>
__device__ __forceinline__ v8f mm_tile(int row0, int col0, int K, FA la, FB lb) {
  int lane = threadIdx.x & 31;
  int r  = lane & 15;
  int kb = (lane >> 4) << 1;
  v8f acc = {};
  for (int k0 = 0; k0 < K; k0 += 4) {
    v2f a, b;
    a.x = la(row0 + r, k0 + kb);
    a.y = la(row0 + r, k0 + kb + 1);
    b.x = lb(k0 + kb,     col0 + r);
    b.y = lb(k0 + kb + 1, col0 + r);
    acc = wmma4(a, b, acc);
  }
  return acc;
}

// ---------------- Stage kernels ----------------

// Twiddle E[i,k] = exp(-2*pi*I*i*k/M) and spectral weights W_c[k1,k2] (FFT-bin order),
// with all constants (deconv^2, mult, scale, 1/M^2 Parseval factor) folded in.
__global__ void k_tables(float* __restrict__ Ere, float* __restrict__ Eim,
                         float* __restrict__ W0,  float* __restrict__ W1,
                         const float* s0, const float* s1,
                         const float* a0, const float* a1) {
  int t = blockIdx.x * blockDim.x + threadIdx.x;
  if (t >= MP * MP) return;
  int i = t / MP, k = t % MP;
  float er = 0.f, ei = 0.f, w0 = 0.f, w1 = 0.f;
  if (i < MM && k < MM) {
    int m = (i * k) % MM;                       // keep angle small & exact
    float th = (2.0f * PI_F / (float)MM) * (float)m;
    er = cosf(th);
    ei = -sinf(th);
    // frequency of FFT bin (L = 2*pi => 2*pi/L = 1)
    float f1 = (float)((i <= MM / 2) ? i : i - MM);
    float f2 = (float)((k <= MM / 2) ? k : k - MM);
    float k2 = f1 * f1 + f2 * f2;
    float dec = (PI_F / TAU_F) * expf(TAU_F * k2);
    // scale = 1/(M^2 * 4*pi^2 * 2); extra 1/M^2 from Parseval
    float m2   = (float)MM * (float)MM;
    float coef = 1.0f / (8.0f * PI_F * PI_F * m2 * m2);
    float base = coef * dec * dec * 4.0f * PI_F;
    float sh0 = s0[0], sh1 = s1[0];             // MU0 = MU1 = 1
    w0 = base * a0[0] / (k2 + sh0 * sh0);
    w1 = base * a1[0] / (k2 + sh1 * sh1);
  }
  Ere[t] = er; Eim[t] = ei; W0[t] = w0; W1[t] = w1;
}

// Separable periodic-Gaussian spread: Gx[pt,i], Gy[pt,i], zero-padded to MP.
__global__ void k_spread(const float* __restrict__ x,
                         float* __restrict__ Gx, float* __restrict__ Gy) {
  int t = blockIdx.x * blockDim.x + threadIdx.x;
  if (t >= NPTS * MP) return;
  int pt = t / MP, i = t % MP;
  float gx = 0.f, gy = 0.f;
  if (i < MM) {
    float xi  = (L_F / (float)MM) * (float)i;
    float inv = 1.0f / (4.0f * TAU_F);
    float d0 = x[2 * pt + 0] - xi;
    float d1 = x[2 * pt + 1] - xi;
    gx = expf(-d0 * d0 * inv) + expf(-(d0 - L_F) * (d0 - L_F) * inv)
       + expf(-(d0 + L_F) * (d0 + L_F) * inv);
    gy = expf(-d1 * d1 * inv) + expf(-(d1 - L_F) * (d1 - L_F) * inv)
       + expf(-(d1 + L_F) * (d1 + L_F) * inv);
  }
  Gx[t] = gx; Gy[t] = gy;
}

// 1-D DFT as GEMM: F = G * E.  z: 0=Gx*Ere, 1=Gx*Eim, 2=Gy*Ere, 3=Gy*Eim
__global__ void k_dft(const float* __restrict__ Gx, const float* __restrict__ Gy,
                      const float* __restrict__ Ere, const float* __restrict__ Eim,
                      float* __restrict__ FxRe, float* __restrict__ FxIm,
                      float* __restrict__ FyRe, float* __restrict__ FyIm) {
  int z = blockIdx.z;
  const float* G = (z < 2) ? Gx : Gy;
  const float* E = (z & 1) ? Eim : Ere;
  float* D = (z == 0) ? FxRe : (z == 1) ? FxIm : (z == 2) ? FyRe : FyIm;
  int row0 = blockIdx.x * 16, col0 = blockIdx.y * 16;
  v8f acc = mm_tile(row0, col0, MP,
                    [&](int r, int k) { return G[r * MP + k]; },
                    [&](int k, int c) { return E[k * MP + c]; });
  store_tile(D, MP, row0, col0, acc);
}

// Power spectra |Fx|^2, |Fy|^2
__global__ void k_sqmag(const float* __restrict__ FxRe, const float* __restrict__ FxIm,
                        const float* __restrict__ FyRe, const float* __restrict__ FyIm,
                        float* __restrict__ Ax, float* __restrict__ Ay) {
  int t = blockIdx.x * blockDim.x + threadIdx.x;
  if (t >= NPTS * MP) return;
  Ax[t] = FxRe[t] * FxRe[t] + FxIm[t] * FxIm[t];
  Ay[t] = FyRe[t] * FyRe[t] + FyIm[t] * FyIm[t];
}

// Per-batch complex Gram: S[k1,k2] = sum_q Fx[q,k1] * Fy[q,k2]   (K = 512)
__global__ void k_smat(const float* __restrict__ FxRe, const float* __restrict__ FxIm,
                       const float* __restrict__ FyRe, const float* __restrict__ FyIm,
                       float* __restrict__ SRe, float* __restrict__ SIm) {
  int b = blockIdx.z;
  int row0 = blockIdx.x * 16, col0 = blockIdx.y * 16;
  int lane = threadIdx.x & 31, r = lane & 15, kb = (lane >> 4) << 1;
  const float* fxr = FxRe + b * NPP * MP;
  const float* fxi = FxIm + b * NPP * MP;
  const float* fyr = FyRe + b * NPP * MP;
  const float* fyi = FyIm + b * NPP * MP;
  v8f accRe = {}, accIm = {};
  for (int k0 = 0; k0 < NPP; k0 += 4) {
    int q0 = k0 + kb, q1 = q0 + 1;
    v2f aRe = { fxr[q0 * MP + row0 + r], fxr[q1 * MP + row0 + r] };
    v2f aIm = { fxi[q0 * MP + row0 + r], fxi[q1 * MP + row0 + r] };
    v2f aImN = { -aIm.x, -aIm.y };
    v2f bRe = { fyr[q0 * MP + col0 + r], fyr[q1 * MP + col0 + r] };
    v2f bIm = { fyi[q0 * MP + col0 + r], fyi[q1 * MP + col0 + r] };
    accRe = wmma4(aRe,  bRe, accRe);
    accRe = wmma4(aImN, bIm, accRe);   // - Im*Im
    accIm = wmma4(aRe,  bIm, accIm);
    accIm = wmma4(aIm,  bRe, accIm);
  }
  store_tile(SRe + b * MP * MP, MP, row0, col0, accRe);
  store_tile(SIm + b * MP * MP, MP, row0, col0, accIm);
}

// U = (W o S) * conj(Fy)^T  -> URe/UIm [MP x NPP] per batch (W o S fused into A-loads)
__global__ void k_umat(const float* __restrict__ W,
                       const float* __restrict__ SRe, const float* __restrict__ SIm,
                       const float* __restrict__ FyRe, const float* __restrict__ FyIm,
                       float* __restrict__ URe, float* __restrict__ UIm) {
  int b = blockIdx.z;
  int row0 = blockIdx.x * 16;   // k1
  int col0 = blockIdx.y * 16;   // p
  int lane = threadIdx.x & 31, r = lane & 15, kb = (lane >> 4) << 1;
  const float* sre = SRe + b * MP * MP;
  const float* sim = SIm + b * MP * MP;
  const float* fyr = FyRe + b * NPP * MP;
  const float* fyi = FyIm + b * NPP * MP;
  v8f accRe = {}, accIm = {};
  for (int k0 = 0; k0 < MP; k0 += 4) {
    int k2a = k0 + kb, k2b = k2a + 1;
    int ra = (row0 + r) * MP;
    float w0 = W[ra + k2a], w1 = W[ra + k2b];
    v2f tRe = { w0 * sre[ra + k2a], w1 * sre[ra + k2b] };
    v2f tIm = { w0 * sim[ra + k2a], w1 * sim[ra + k2b] };
    v2f tReN = { -tRe.x, -tRe.y };
    int pb = (col0 + r) * MP;
    v2f bRe = { fyr[pb + k2a], fyr[pb + k2b] };
    v2f bIm = { fyi[pb + k2a], fyi[pb + k2b] };
    accRe = wmma4(tRe,  bRe, accRe);   // Re{T*conj(fy)} = TRe*FyRe + TIm*FyIm
    accRe = wmma4(tIm,  bIm, accRe);
    accIm = wmma4(tIm,  bRe, accIm);   // Im{T*conj(fy)} = TIm*FyRe - TRe*FyIm
    accIm = wmma4(tReN, bIm, accIm);
  }
  store_tile(URe + b * MP * NPP, NPP, row0, col0, accRe);
  store_tile(UIm + b * MP * NPP, NPP, row0, col0, accIm);
}

// V = W * Ay^T  -> [MP x NPP] per batch (diag term)
__global__ void k_vmat(const float* __restrict__ W, const float* __restrict__ Ay,
                       float* __restrict__ V) {
  int b = blockIdx.z;
  int row0 = blockIdx.x * 16, col0 = blockIdx.y * 16;
  const float* ay = Ay + b * NPP * MP;
  v8f acc = mm_tile(row0, col0, MP,
                    [&](int r, int k) { return W[r * MP + k]; },
                    [&](int k, int c) { return ay[c * MP + k]; });
  store_tile(V + b * MP * NPP, NPP, row0, col0, acc);
}

// Per-point reduction: out[pt,ch] = Re(conj(fx)·U[:,p]) - Ax·V[:,p]
__global__ void k_reduce(const float* __restrict__ FxRe, const float* __restrict__ FxIm,
                         const float* __restrict__ Ax,
                         const float* __restrict__ URe, const float* __restrict__ UIm,
                         const float* __restrict__ V,
                         float* __restrict__ out, int ch) {
  int pt = blockIdx.x;                 // 0..2047
  int b = pt >> 9, p = pt & (NPP - 1);
  int lane = threadIdx.x;
  float tot = 0.f, dia = 0.f;
  for (int k = lane; k < MP; k += 32) {
    int uc = (b * MP + k) * NPP + p;
    tot += FxRe[pt * MP + k] * URe[uc] + FxIm[pt * MP + k] * UIm[uc];
    dia += Ax[pt * MP + k] * V[uc];
  }
#pragma unroll
  for (int off = 16; off > 0; off >>= 1) {
    tot += __shfl_down(tot, off);
    dia += __shfl_down(dia, off);
  }
  if (lane == 0) out[pt * 2 + ch] = tot - dia;
}

// ---------------- Launch ----------------

extern "C" void kernel_launch(void* const* d_in, const int* in_sizes, int n_in,
                              void* d_out, int out_size, void* d_ws, size_t ws_size,
                              hipStream_t stream) {
  (void)in_sizes; (void)n_in; (void)out_size; (void)ws_size;
  const float* x  = (const float*)d_in[0];
  const float* s0 = (const float*)d_in[1];
  const float* s1 = (const float*)d_in[2];
  const float* a0 = (const float*)d_in[3];
  const float* a1 = (const float*)d_in[4];
  float* out = (float*)d_out;

  float* w = (float*)d_ws;
  float* Ere  = w; w += MP * MP;
  float* Eim  = w; w += MP * MP;
  float* W0   = w; w += MP * MP;
  float* W1   = w; w += MP * MP;
  float* Gx   = w; w += NPTS * MP;
  float* Gy   = w; w += NPTS * MP;
  float* FxRe = w; w += NPTS * MP;
  float* FxIm = w; w += NPTS * MP;
  float* FyRe = w; w += NPTS * MP;
  float* FyIm = w; w += NPTS * MP;
  float* Ax   = w; w += NPTS * MP;
  float* Ay   = w; w += NPTS * MP;
  float* SRe  = w; w += NB * MP * MP;
  float* SIm  = w; w += NB * MP * MP;
  float* URe  = w; w += NB * MP * NPP;
  float* UIm  = w; w += NB * MP * NPP;
  float* V    = w; w += NB * MP * NPP;   // total ~14 MB of f32 scratch

  k_tables<<<(MP * MP + 255) / 256, 256, 0, stream>>>(Ere, Eim, W0, W1, s0, s1, a0, a1);
  k_spread<<<(NPTS * MP + 255) / 256, 256, 0, stream>>>(x, Gx, Gy);
  k_dft<<<dim3(NPTS / 16, MP / 16, 4), 32, 0, stream>>>(Gx, Gy, Ere, Eim,
                                                        FxRe, FxIm, FyRe, FyIm);
  k_sqmag<<<(NPTS * MP + 255) / 256, 256, 0, stream>>>(FxRe, FxIm, FyRe, FyIm, Ax, Ay);
  k_smat<<<dim3(MP / 16, MP / 16, NB), 32, 0, stream>>>(FxRe, FxIm, FyRe, FyIm, SRe, SIm);

  for (int c = 0; c < 2; ++c) {
    const float* Wc = c ? W1 : W0;
    k_umat<<<dim3(MP / 16, NPP / 16, NB), 32, 0, stream>>>(Wc, SRe, SIm, FyRe, FyIm, URe, UIm);
    k_vmat<<<dim3(MP / 16, NPP / 16, NB), 32, 0, stream>>>(Wc, Ay, V);
    k_reduce<<<NPTS, 32, 0, stream>>>(FxRe, FxIm, Ax, URe, UIm, V, out, c);
  }
}